// GraphConvolution_37787122270314
// MI455X (gfx1250) — compile-verified
//
#include <hip/hip_runtime.h>

typedef __attribute__((ext_vector_type(2))) float v2f;
typedef __attribute__((ext_vector_type(8))) float v8f;

#define DCOL 512          // D_IN == D_OUT == 512
#define EPB  64           // edges per block in the SpMM phase

// ---------------------------------------------------------------------------
// Zero-fill kernel (d_out is poisoned by the harness before timing)
// ---------------------------------------------------------------------------
__global__ void gcn_zero_f32(float* __restrict__ p, int n) {
    int i = blockIdx.x * blockDim.x + threadIdx.x;
    if (i < n) p[i] = 0.0f;
}

// ---------------------------------------------------------------------------
// Phase 1: support = inputs @ weight, fp32 via V_WMMA_F32_16X16X4_F32.
// grid.x = nNodes/16 blocks; 256 threads = 8 waves; each wave owns a
// 16-row x 64-col slab (4 accumulators of 16x16), block covers 16 x 512.
// ---------------------------------------------------------------------------
__global__ __launch_bounds__(256)
void gcn_gemm_wmma(const float* __restrict__ A,   // [nNodes, 512] row-major
                   const float* __restrict__ W,   // [512, 512]    row-major
                   float* __restrict__ S,         // [nNodes, 512] support out
                   int nNodes) {
    const int lane  = threadIdx.x & 31;
    const int wave  = threadIdx.x >> 5;
    const int mBase = blockIdx.x * 16;
    const int nBase = wave * 64;

    const int half = lane >> 4;      // 0: K pair {0,1}; 1: K pair {2,3}
    const int l16  = lane & 15;      // M index (A) / N index (B,C,D)
    const int kOff = half * 2;

    v8f acc0 = {}, acc1 = {}, acc2 = {}, acc3 = {};

    int arowIdx = mBase + l16;
    if (arowIdx >= nNodes) arowIdx = nNodes - 1;   // clamp (nNodes%16==0 anyway)
    const float* arow = A + (size_t)arowIdx * DCOL;

    for (int k = 0; k < DCOL; k += 4) {
        // A fragment: lane holds A[m][k+kOff], A[m][k+kOff+1]
        v2f a;
        a.x = arow[k + kOff];
        a.y = arow[k + kOff + 1];

        // B fragments: lane holds W[k+kOff][n], W[k+kOff+1][n]
        const float* w0 = W + (size_t)(k + kOff)     * DCOL + nBase + l16;
        const float* w1 = W + (size_t)(k + kOff + 1) * DCOL + nBase + l16;
        v2f b0, b1, b2, b3;
        b0.x = w0[0];   b0.y = w1[0];
        b1.x = w0[16];  b1.y = w1[16];
        b2.x = w0[32];  b2.y = w1[32];
        b3.x = w0[48];  b3.y = w1[48];

        acc0 = __builtin_amdgcn_wmma_f32_16x16x4_f32(false, a, false, b0,
                                                     (short)0, acc0, false, false);
        acc1 = __builtin_amdgcn_wmma_f32_16x16x4_f32(false, a, false, b1,
                                                     (short)0, acc1, false, false);
        acc2 = __builtin_amdgcn_wmma_f32_16x16x4_f32(false, a, false, b2,
                                                     (short)0, acc2, false, false);
        acc3 = __builtin_amdgcn_wmma_f32_16x16x4_f32(false, a, false, b3,
                                                     (short)0, acc3, false, false);
    }

    // D layout: VGPR r -> row mBase + r + 8*half, col nBase + (tile*16) + l16
#pragma unroll
    for (int r = 0; r < 8; ++r) {
        int row = mBase + r + 8 * half;
        if (row < nNodes) {
            float* srow = S + (size_t)row * DCOL + nBase + l16;
            srow[0]  = acc0[r];
            srow[16] = acc1[r];
            srow[32] = acc2[r];
            srow[48] = acc3[r];
        }
    }
}

// ---------------------------------------------------------------------------
// Phase 2: out[adj_rows[e]] += support[adj_cols[e]] * adj_vals[e]
// adj_rows is SORTED: accumulate a row-run in registers, flush with
// atomicAdd only on row change / block boundary (~32 edges per row).
// 128 threads x float4 = 512 columns per block; EPB edges per block.
// ---------------------------------------------------------------------------
__global__ __launch_bounds__(128)
void gcn_spmm(const float* __restrict__ S,
              const int*   __restrict__ rows,
              const int*   __restrict__ cols,
              const float* __restrict__ vals,
              float* __restrict__ out, int nE) {
    const int c0 = threadIdx.x * 4;
    int e    = blockIdx.x * EPB;
    int eEnd = e + EPB;
    if (eEnd > nE) eEnd = nE;
    if (e >= eEnd) return;

    float4 acc = make_float4(0.f, 0.f, 0.f, 0.f);

    for (; e < eEnd; ++e) {
        const int   row = rows[e];
        const int   col = cols[e];
        const float val = vals[e];

        // prefetch a future gather target into cache (global_prefetch_b8)
        if (e + 8 < nE)
            __builtin_prefetch(S + (size_t)cols[e + 8] * DCOL + c0, 0, 1);

        const float4 sv = *(const float4*)(S + (size_t)col * DCOL + c0);
        acc.x += sv.x * val;
        acc.y += sv.y * val;
        acc.z += sv.z * val;
        acc.w += sv.w * val;

        const bool flush = (e + 1 == eEnd) || (rows[e + 1] != row);
        if (flush) {
            float* o = out + (size_t)row * DCOL + c0;
            atomicAdd(o + 0, acc.x);
            atomicAdd(o + 1, acc.y);
            atomicAdd(o + 2, acc.z);
            atomicAdd(o + 3, acc.w);
            acc = make_float4(0.f, 0.f, 0.f, 0.f);
        }
    }
}

// ---------------------------------------------------------------------------
// Launch
// ---------------------------------------------------------------------------
extern "C" void kernel_launch(void* const* d_in, const int* in_sizes, int n_in,
                              void* d_out, int out_size, void* d_ws, size_t ws_size,
                              hipStream_t stream) {
    const float* inputs   = (const float*)d_in[0];   // [N_NODES, 512]
    const float* weight   = (const float*)d_in[1];   // [512, 512]
    const int*   adj_rows = (const int*)  d_in[2];   // [E] sorted
    const int*   adj_cols = (const int*)  d_in[3];   // [E]
    const float* adj_vals = (const float*)d_in[4];   // [E]
    float*       out      = (float*)d_out;           // [N_NODES, 512]
    float*       support  = (float*)d_ws;            // [N_NODES, 512] scratch

    const int nNodes = in_sizes[0] / DCOL;           // 50000
    const int nE     = in_sizes[2];                  // 1600000

    // zero the output (harness poisons it)
    {
        int n = out_size;
        int blocks = (n + 255) / 256;
        gcn_zero_f32<<<blocks, 256, 0, stream>>>(out, n);
    }

    // Phase 1: fp32 WMMA GEMM into workspace
    {
        int mTiles = (nNodes + 15) / 16;             // 3125
        gcn_gemm_wmma<<<mTiles, 256, 0, stream>>>(inputs, weight, support, nNodes);
    }

    // Phase 2: gather + sorted segment-sum
    {
        int blocks = (nE + EPB - 1) / EPB;           // 25000
        gcn_spmm<<<blocks, 128, 0, stream>>>(support, adj_rows, adj_cols,
                                             adj_vals, out, nE);
    }
}